// SMCL_19104014533275
// MI455X (gfx1250) — compile-verified
//
#include <hip/hip_runtime.h>
#include <stdint.h>

// Hard-negative contrastive loss for MI455X (gfx1250).
// Memory-bound: 256 MiB single pass -> ~11.5 us floor @ 23.3 TB/s.
// Strategy: one workgroup per row; async-copy the 64 KB row into LDS
// (CDNA5 global_load_async_to_lds_b128, tracked by ASYNCcnt), then an exact
// LDS-resident radix/binary select for the k-th largest key, then one
// deterministic pass for sum((v+1)^2) above threshold.

#define NT    256      // 8 wave32 per block
#define N_MAX 16384    // row length (fits 64 KB LDS)
#define CAP   2048     // candidate buffer (8 KB)

// order-preserving f32 -> u32 key (larger key == larger float)
__device__ __forceinline__ unsigned f2k(float f) {
  unsigned u = __float_as_uint(f);
  return (u & 0x80000000u) ? ~u : (u | 0x80000000u);
}
__device__ __forceinline__ float k2f(unsigned kk) {
  unsigned u = (kk & 0x80000000u) ? (kk & 0x7fffffffu) : ~kk;
  return __uint_as_float(u);
}

__global__ __launch_bounds__(NT) void topk_loss_rows(
    const float* __restrict__ inputs, const int* __restrict__ targets,
    float* __restrict__ row_loss, int n, int k, float delta) {
  __shared__ __align__(16) float vals[N_MAX];   // raw row
  __shared__ __align__(16) float cand[CAP];     // compacted survivors
  __shared__ float s_red[NT];
  __shared__ unsigned s_cnt, s_candCnt, s_T, s_cnt2;
  __shared__ float s_pos;

  const int tid  = threadIdx.x;
  const int lane = tid & 31;
  const int row  = blockIdx.x;
  const float* rp = inputs + (size_t)row * (size_t)n;

  // ---- Phase 0: async-copy row -> LDS (b128 per op, coalesced) ----------
  for (int i = tid * 4; i < n; i += NT * 4) {
    unsigned dst = (unsigned)(uintptr_t)(&vals[i]);   // low 32 bits = LDS offset
    const float* src = rp + i;
    asm volatile("global_load_async_to_lds_b128 %0, %1, off"
                 :: "v"(dst), "v"(src) : "memory");
  }
  asm volatile("s_wait_asynccnt 0" ::: "memory");     // my wave's copies done
  __syncthreads();                                    // everyone's copies done

  // save positive logit, mask it out (key(-inf) is the global minimum key)
  if (tid == 0) {
    int t  = targets[row];
    s_pos  = vals[t];
    vals[t] = -__builtin_inff();
  }
  __syncthreads();

  // ---- Phase A: block-wide binary search on key space --------------------
  // invariant: count(key > lo) >= k  and  count(key > hi) < k
  unsigned lo = 0u, hi = 0xffffffffu;
  unsigned flo = (unsigned)n;                 // upper bound on count(key > lo)
  while (flo > CAP && (hi - lo) > 1u) {
    unsigned mid = lo + ((hi - lo) >> 1);
    if (tid == 0) s_cnt = 0u;
    __syncthreads();
    unsigned c = 0u;
    for (int e = tid; e < n; e += NT) c += (f2k(vals[e]) > mid) ? 1u : 0u;
    atomicAdd(&s_cnt, c);
    __syncthreads();
    unsigned tot = s_cnt;
    __syncthreads();                          // s_cnt reused next round
    if (tot >= (unsigned)k) { lo = mid; flo = tot; } else { hi = mid; }
  }

  // ---- Phase B+C: compact survivors, wave 0 finishes the search ----------
  if (flo <= CAP && (hi - lo) > 1u) {         // block-uniform condition
    if (tid == 0) s_candCnt = 0u;
    __syncthreads();
    for (int e = tid; e < n; e += NT) {       // n % NT == 0 for this shape
      float v = vals[e];
      bool p = f2k(v) > lo;
      unsigned long long b = __ballot(p);
      unsigned prefix = (unsigned)__popcll(b & ((1ull << lane) - 1ull));
      unsigned cntw   = (unsigned)__popcll(b);
      unsigned base = 0u;
      if (lane == 0 && cntw) base = atomicAdd(&s_candCnt, cntw);
      base = __shfl(base, 0);
      if (p) cand[base + prefix] = v;
    }
    __syncthreads();
    if (tid < 32) {                           // wave 0 only: no barriers needed
      const unsigned cc = s_candCnt;          // <= CAP, == count(key > lo)
      while ((hi - lo) > 1u) {
        unsigned mid = lo + ((hi - lo) >> 1);
        unsigned c = 0u;
        for (unsigned i = (unsigned)lane; i < cc; i += 32u)
          c += (f2k(cand[i]) > mid) ? 1u : 0u;
        for (int o = 16; o > 0; o >>= 1) c += __shfl_down(c, o);
        c = __shfl(c, 0);
        if (c >= (unsigned)k) lo = mid; else hi = mid;
      }
    }
  }
  if (tid == 0) s_T = hi;                     // hi == k-th largest key, exists in data
  __syncthreads();
  const unsigned T = s_T;

  // ---- Phase D: deterministic sum of (v+1)^2 strictly above T ------------
  float sum = 0.f;
  unsigned cnt = 0u;
  for (int e = tid; e < n; e += NT) {
    float v = vals[e];
    if (f2k(v) > T) { float w = v + 1.f; sum += w * w; cnt++; }
  }
  if (tid == 0) s_cnt2 = 0u;
  s_red[tid] = sum;
  __syncthreads();
  atomicAdd(&s_cnt2, cnt);                    // integer: order-independent
  for (int o = NT / 2; o > 0; o >>= 1) {      // deterministic tree for the float
    if (tid < o) s_red[tid] += s_red[tid + o];
    __syncthreads();
  }
  if (tid == 0) {
    float tv  = k2f(T);
    float w   = tv + 1.f;
    float rem = (float)(k - (int)s_cnt2);     // ties at T: cnt2 < k always
    float total = s_red[0] + rem * w * w;
    float pd  = s_pos - 1.f;
    row_loss[row] = delta * pd * pd + total / (float)k;
  }
}

__global__ __launch_bounds__(NT) void reduce_mean(
    const float* __restrict__ rl, float* __restrict__ out, int m) {
  __shared__ float s[NT];
  float a = 0.f;
  for (int i = threadIdx.x; i < m; i += NT) a += rl[i];
  s[threadIdx.x] = a;
  __syncthreads();
  for (int o = NT / 2; o > 0; o >>= 1) {
    if (threadIdx.x < o) s[threadIdx.x] += s[threadIdx.x + o];
    __syncthreads();
  }
  if (threadIdx.x == 0) out[0] = s[0] / (float)m;
}

extern "C" void kernel_launch(void* const* d_in, const int* in_sizes, int n_in,
                              void* d_out, int out_size, void* d_ws, size_t ws_size,
                              hipStream_t stream) {
  const float* inputs  = (const float*)d_in[0];
  const int*   targets = (const int*)d_in[1];
  const int m = in_sizes[1];
  const int n = in_sizes[0] / m;               // 16384
  int k = (int)(0.01 * (double)(n - 1));       // == int(GAMMA * num_neg) = 163
  if (k < 1) k = 1;

  float* row_loss = (float*)d_ws;              // m floats of scratch
  topk_loss_rows<<<m, NT, 0, stream>>>(inputs, targets, row_loss, n, k, 5.0f);
  reduce_mean<<<1, NT, 0, stream>>>(row_loss, (float*)d_out, m);
}